// WaveletModule_41644002902514
// MI455X (gfx1250) — compile-verified
//
#include <hip/hip_runtime.h>
#include <math.h>

typedef __attribute__((ext_vector_type(2))) float v2f;
typedef __attribute__((ext_vector_type(8))) float v8f;

#define T_LEN   2000
#define K_LEN   1601
#define K_PAD   1632       // zero-padded taps; 408 K-steps of 4 (divisible by 8)
#define NF      100
#define NF_PAD  112
#define PADL    800
#define XP_LEN  3632       // covers max index 1969 + (K_PAD-1) = 3600
#define NROWS   256        // 8 batches * 32 channels
#define NTAU    64         // 32 target steps * 2 (i0,i1)
#define NBANDS  5
#define OUT_N   (8*5*32*32)

// workspace layout (float offsets)
#define OFF_XPAD 0
#define OFF_WR   (NROWS * XP_LEN)
#define OFF_WI   (OFF_WR + NF_PAD * K_PAD)
#define OFF_BAND (OFF_WI + NF_PAD * K_PAD)

// ---------------------------------------------------------------- zero band buffer
__global__ void k_zero(float* __restrict__ p, int n) {
    int i = blockIdx.x * blockDim.x + threadIdx.x;
    if (i < n) p[i] = 0.0f;
}

// ---------------------------------------------------------------- pad weights: (100,1601) -> (112,1632) zero-filled
__global__ void k_prepw(const float* __restrict__ wr, const float* __restrict__ wi,
                        float* __restrict__ WR, float* __restrict__ WI) {
    int i = blockIdx.x * blockDim.x + threadIdx.x;
    if (i >= NF_PAD * K_PAD) return;
    int f = i / K_PAD;
    int k = i - f * K_PAD;
    float r = 0.0f, m = 0.0f;
    if (f < NF && k < K_LEN) {
        r = wr[f * K_LEN + k];
        m = wi[f * K_LEN + k];
    }
    WR[i] = r;
    WI[i] = m;
}

// ---------------------------------------------------------------- standardize + clip + edge-pad each row
__global__ __launch_bounds__(256) void k_prepx(const float* __restrict__ eeg,
                                               float* __restrict__ xpad) {
    int row = blockIdx.x;
    int tid = threadIdx.x;
    const float* x = eeg + row * T_LEN;

    float s = 0.0f, ss = 0.0f;
    for (int i = tid; i < T_LEN; i += 256) {
        float v = x[i];
        s  += v;
        ss += v * v;
    }
    __shared__ float rs[256], rss[256];
    rs[tid] = s; rss[tid] = ss;
    __syncthreads();
    for (int off = 128; off > 0; off >>= 1) {
        if (tid < off) { rs[tid] += rs[tid + off]; rss[tid] += rss[tid + off]; }
        __syncthreads();
    }
    float mean = rs[0] * (1.0f / T_LEN);
    float var  = (rss[0] - (float)T_LEN * mean * mean) * (1.0f / (T_LEN - 1));
    float lim  = 6.0f * sqrtf(fmaxf(var, 0.0f));

    float* xp = xpad + row * XP_LEN;
    for (int i = tid; i < XP_LEN; i += 256) {
        int src = i - PADL;
        src = src < 0 ? 0 : (src > T_LEN - 1 ? T_LEN - 1 : src);   // 'edge' pad == clamp
        float v = x[src] - mean;
        v = fminf(fmaxf(v, -lim), lim);
        xp[i] = v;                    // tail beyond 3600 only meets zero weights
    }
}

// ---------------------------------------------------------------- WMMA CWT + power + band reduce
// One block per row, 7 wave32s. Wave w owns freq-tile mt=w and sweeps all 4
// tau-tiles with one A fragment per K-step: 8 v_wmma per A-load pair.
__global__ __launch_bounds__(224) void k_cwt(const float* __restrict__ xpad,
                                             const float* __restrict__ WR,
                                             const float* __restrict__ WI,
                                             float* __restrict__ band) {
    __shared__ float sx[XP_LEN];
    const int row = blockIdx.x;
    const int tid = threadIdx.x;

    const float* xg = xpad + row * XP_LEN;
    for (int i = tid; i < XP_LEN; i += 224) sx[i] = xg[i];
    __syncthreads();

    const int mt    = tid >> 5;           // 0..6 (freq tile == wave id)
    const int lane  = tid & 31;
    const int mlane = lane & 15;
    const int ksel  = (lane >> 4) << 1;   // lanes 0-15 -> K0,K1 ; lanes 16-31 -> K2,K3

    // A-fragment row for this lane (< 112, zero-padded weights)
    const int f = mt * 16 + mlane;
    const float* ar = WR + f * K_PAD + ksel;   // 8B-aligned (K_PAD%4==0, ksel even)
    const float* ai = WI + f * K_PAD + ksel;

    // conv-output positions for the 4 tau tiles this wave sweeps
    int tu[4];
    #pragma unroll
    for (int nt = 0; nt < 4; ++nt) {
        const int u = nt * 16 + mlane;          // global tau index 0..63
        const int j = u >> 1, r = u & 1;
        const float srcp = ((float)j + 0.5f) * 62.5f - 0.5f;  // positive -> trunc==floor
        tu[nt] = (int)srcp + r;
    }

    v8f accR[4] = { v8f{}, v8f{}, v8f{}, v8f{} };
    v8f accI[4] = { v8f{}, v8f{}, v8f{}, v8f{} };

    for (int kk = 0; kk < K_PAD; kk += 4) {   // 408 iterations: even, unroll-friendly
        const v2f a0 = *(const v2f*)(ar + kk);
        const v2f a1 = *(const v2f*)(ai + kk);
        #pragma unroll
        for (int nt = 0; nt < 4; ++nt) {
            const float* bx = sx + tu[nt] + ksel + kk;
            v2f b;
            b.x = bx[0];
            b.y = bx[1];
            accR[nt] = __builtin_amdgcn_wmma_f32_16x16x4_f32(false, a0, false, b,
                                                             (short)0, accR[nt], false, false);
            accI[nt] = __builtin_amdgcn_wmma_f32_16x16x4_f32(false, a1, false, b,
                                                             (short)0, accI[nt], false, false);
        }
    }

    // power = yr^2 + yi^2, fold in band weights (_BAND_W hard constants).
    // Bands overlap at boundary freqs: ranges inclusive in freq-index space.
    const int   lo[NBANDS] = { 0,  6, 14, 26, 60 };
    const int   hi[NBANDS] = { 6, 14, 26, 60, 98 };
    const float wb[NBANDS] = { 1.0f/7.0f, 1.0f/9.0f, 1.0f/13.0f, 1.0f/35.0f, 1.0f/39.0f };
    const int mbase = (lane < 16) ? 0 : 8;   // C/D layout: VGPR v -> M = v (+8 upper lanes)

    #pragma unroll
    for (int nt = 0; nt < 4; ++nt) {
        float bacc[NBANDS] = { 0.f, 0.f, 0.f, 0.f, 0.f };
        #pragma unroll
        for (int v = 0; v < 8; ++v) {
            int fv = mt * 16 + mbase + v;
            float p = accR[nt][v] * accR[nt][v] + accI[nt][v] * accI[nt][v];
            #pragma unroll
            for (int n = 0; n < NBANDS; ++n)
                if (fv >= lo[n] && fv <= hi[n]) bacc[n] += p * wb[n];
        }
        const int u = nt * 16 + mlane;
        float* bp = band + (row * NBANDS) * NTAU + u;
        #pragma unroll
        for (int n = 0; n < NBANDS; ++n)
            atomicAdd(bp + n * NTAU, bacc[n]);
    }
}

// ---------------------------------------------------------------- interpolate + transpose to (8,5,32,32)
__global__ void k_final(const float* __restrict__ band, float* __restrict__ out) {
    int i = blockIdx.x * blockDim.x + threadIdx.x;
    if (i >= OUT_N) return;
    int c = i & 31;
    int j = (i >> 5) & 31;
    int n = (i >> 10) % 5;
    int b = i / (5 * 32 * 32);
    int row = b * 32 + c;
    float srcp = ((float)j + 0.5f) * 62.5f - 0.5f;
    int   t0   = (int)srcp;
    float w    = srcp - (float)t0;
    const float* bp = band + (row * NBANDS + n) * NTAU + 2 * j;
    out[i] = bp[0] * (1.0f - w) + bp[1] * w;
}

// ----------------------------------------------------------------
extern "C" void kernel_launch(void* const* d_in, const int* in_sizes, int n_in,
                              void* d_out, int out_size, void* d_ws, size_t ws_size,
                              hipStream_t stream) {
    const float* eeg = (const float*)d_in[0];   // (8,32,2000) f32
    const float* wr  = (const float*)d_in[1];   // (100,1,1601) f32
    const float* wi  = (const float*)d_in[2];   // (100,1,1601) f32

    float* ws   = (float*)d_ws;
    float* xpad = ws + OFF_XPAD;
    float* WR   = ws + OFF_WR;
    float* WI   = ws + OFF_WI;
    float* band = ws + OFF_BAND;

    const int nband = NROWS * NBANDS * NTAU;
    k_zero <<<(nband + 255) / 256, 256, 0, stream>>>(band, nband);
    k_prepw<<<(NF_PAD * K_PAD + 255) / 256, 256, 0, stream>>>(wr, wi, WR, WI);
    k_prepx<<<NROWS, 256, 0, stream>>>(eeg, xpad);
    k_cwt  <<<NROWS, 224, 0, stream>>>(xpad, WR, WI, band);
    k_final<<<(OUT_N + 255) / 256, 256, 0, stream>>>(band, (float*)d_out);
}